// MaxoutActLayer_35828617183845
// MI455X (gfx1250) — compile-verified
//
#include <hip/hip_runtime.h>

typedef __attribute__((ext_vector_type(2))) float v2f;
typedef __attribute__((ext_vector_type(4))) float v4f;
typedef __attribute__((ext_vector_type(8))) float v8f;

#define N_PIXELS (8 * 256 * 256)       // B*H*W
#define C_IN     64
#define N_TILES  (N_PIXELS / 16)       // 32768 tiles of 16 pixels
#define OUT_DIM  64

#define THREADS  256                   // 8 wave32s per block
#define BLOCKS   1024
#define N_WAVES  (BLOCKS * THREADS / 32)        // 8192
#define TILES_PER_WAVE (N_TILES / N_WAVES)      // 4 (exact)

__global__ __launch_bounds__(THREADS) void maxout_wmma_kernel(
    const float* __restrict__ x,    // [N_PIXELS, 64]
    const float* __restrict__ Wm,   // [64, 4]
    const float* __restrict__ bias, // [4]
    float* __restrict__ out) {      // [N_PIXELS, 64]
  const int lane = threadIdx.x & 31;
  const int n    = lane & 15;   // A: filter row (valid < 4); B/D: pixel column
  const int h    = lane >> 4;   // half of wave
  const int waveId = (blockIdx.x * THREADS + threadIdx.x) >> 5;

  // ---- Channel permutation: step-pair (2s, 2s+1), lane-half h covers
  // channels c = 8s + 4h + {0,1,2,3}. Summation order is irrelevant for the
  // GEMM, and this makes each lane's per-step-pair x data 16B contiguous.
  // A layout (16x4 f32): lane (n,h) -> row M=n (filter), VGPR j -> K slot.
  v2f wA[16];
#pragma unroll
  for (int k = 0; k < 16; ++k) { wA[k].x = 0.0f; wA[k].y = 0.0f; }
  if (n < 4) {
#pragma unroll
    for (int s = 0; s < 8; ++s) {
      const int c0 = 8 * s + 4 * h;
      wA[2 * s].x     = Wm[(c0 + 0) * 4 + n];
      wA[2 * s].y     = Wm[(c0 + 1) * 4 + n];
      wA[2 * s + 1].x = Wm[(c0 + 2) * 4 + n];
      wA[2 * s + 1].y = Wm[(c0 + 3) * 4 + n];
    }
  }

  // Bias per D row: acc VGPR r, half h -> filter M = r + 8h.
  // Real filters (M<4, half 0) get bias; padded filters init 0 (relu-max safe).
  float binit[8];
#pragma unroll
  for (int r = 0; r < 8; ++r) binit[r] = 0.0f;
  if (h == 0) {
#pragma unroll
    for (int r = 0; r < 4; ++r) binit[r] = bias[r];
  }

  // ---- 4 tiles per wave, compile-time trip count (EXEC all-ones throughout).
#pragma unroll 1
  for (int i = 0; i < TILES_PER_WAVE; ++i) {
    const long t   = (long)waveId + (long)i * N_WAVES;
    const long pix = t * 16 + n;
    const float* xbase = x + pix * C_IN + 4 * h;

    // One clause of 8 NT b128 loads: the whole 4KB tile into registers.
    v4f q[8];
#pragma unroll
    for (int s = 0; s < 8; ++s) {
      q[s] = __builtin_nontemporal_load((const v4f*)(xbase + 8 * s));
    }

    v8f acc;
#pragma unroll
    for (int r = 0; r < 8; ++r) acc[r] = binit[r];

#pragma unroll
    for (int s = 0; s < 8; ++s) {
      v2f b0; b0.x = q[s].x; b0.y = q[s].y;   // B: K slots for step 2s
      acc = __builtin_amdgcn_wmma_f32_16x16x4_f32(
          false, wA[2 * s], false, b0, (short)0, acc, false, false);
      v2f b1; b1.x = q[s].z; b1.y = q[s].w;   // B: K slots for step 2s+1
      acc = __builtin_amdgcn_wmma_f32_16x16x4_f32(
          false, wA[2 * s + 1], false, b1, (short)0, acc, false, false);
    }

    // ---- Maxout in-lane: filters 0..3 live in acc[0..3] (half 0); half 1
    // holds zero-padded filters -> computes 0. Real max >= 0, so one xor-16
    // shuffle gives every lane its pixel's maxout value.
    float v = fmaxf(fmaxf(acc[0], acc[1]), fmaxf(acc[2], acc[3]));
    v = fmaxf(v, 0.0f);
    v = fmaxf(v, __shfl_xor(v, 16, 32));

    // ---- Broadcast to 64 output channels: 2 lanes per pixel, 8 float4 each
    // (8 NT b128 stores per tile, 4KB written = 4KB read).
    v4f o; o.x = v; o.y = v; o.z = v; o.w = v;
    float* orow = out + pix * OUT_DIM + h * 4;
#pragma unroll
    for (int j = 0; j < 8; ++j) {
      __builtin_nontemporal_store(o, (v4f*)(orow + j * 8));
    }
  }
}

extern "C" void kernel_launch(void* const* d_in, const int* in_sizes, int n_in,
                              void* d_out, int out_size, void* d_ws, size_t ws_size,
                              hipStream_t stream) {
  (void)in_sizes; (void)n_in; (void)out_size; (void)d_ws; (void)ws_size;
  const float* x    = (const float*)d_in[0];
  const float* Wm   = (const float*)d_in[1];
  const float* bias = (const float*)d_in[2];
  float* out        = (float*)d_out;

  hipLaunchKernelGGL(maxout_wmma_kernel, dim3(BLOCKS), dim3(THREADS), 0, stream,
                     x, Wm, bias, out);
}